// VirtualGeometryModule_51402168599382
// MI455X (gfx1250) — compile-verified
//
#include <hip/hip_runtime.h>
#include <math.h>
#include <stdint.h>

#define EPS_LEN   1e-20f
#define EPS_DENOM 1e-06f

#define TILE 256              // faces per tile == blockDim.x
#define TILE_DW (TILE * 3)    // 768 dwords per index-tile

struct __align__(4) V3f { float x, y, z; };
struct __align__(8) V2f { float x, y; };

// ---- CDNA5-specific primitives (inline asm, probe-validated syntax) --------

// Native fp32 global atomic add, device scope (RMW at L2; the 48MB of
// accumulators are fully L2-resident on MI455X's 192MB L2).
__device__ __forceinline__ void gatomic_add_f32(float* p, float v) {
    asm volatile("global_atomic_add_f32 %0, %1, off scope:SCOPE_DEV"
                 :: "v"(p), "v"(v)
                 : "memory");
}

// Async copy global->LDS (ASYNCcnt-tracked, bypasses the VGPR load path).
__device__ __forceinline__ void async_ld_b32(uint32_t lds_off, const int* gaddr) {
    asm volatile("global_load_async_to_lds_b32 %0, %1, off"
                 :: "v"(lds_off), "v"(gaddr)
                 : "memory");
}
__device__ __forceinline__ void wait_async6() {   // previous tile retired
    asm volatile("s_wait_asynccnt 0x6" ::: "memory");
}
__device__ __forceinline__ void wait_async0() {   // all async retired
    asm volatile("s_wait_asynccnt 0x0" ::: "memory");
}

// Generic pointer to an LDS object: low 32 bits are the LDS byte offset
// (flat LDS aperture truncation rule, ISA 10.2).
__device__ __forceinline__ uint32_t lds_off_of(const void* p) {
    return (uint32_t)(uintptr_t)p;
}

// ---------------------------------------------------------------------------

__global__ __launch_bounds__(256) void vg_zero_kernel(float4* __restrict__ out4,
                                                      int n4,
                                                      float* __restrict__ out,
                                                      int n) {
    int i = blockIdx.x * blockDim.x + threadIdx.x;
    if (i < n4) out4[i] = make_float4(0.f, 0.f, 0.f, 0.f);
    if (i == 0) {
        for (int k = n4 * 4; k < n; ++k) out[k] = 0.f;
    }
}

// Face-parallel pass. Persistent blocks grid-stride over 256-face tiles.
// Index streams (faces/uv_faces) are staged into double-buffered LDS via the
// async engine; positions/texcoords are random-gathered (L2-resident);
// results scattered with hardware f32 atomics.
__global__ __launch_bounds__(256) void vg_face_kernel(const float* __restrict__ positions,
                                                      const int*   __restrict__ faces,
                                                      const float* __restrict__ texcoords,
                                                      const int*   __restrict__ uv_faces,
                                                      float* __restrict__ nrm_accum,
                                                      float* __restrict__ tan_accum,
                                                      int F,
                                                      int numTiles) {
    __shared__ int facesLDS[2][TILE_DW];
    __shared__ int uvLDS[2][TILE_DW];

    const int t = threadIdx.x;
    const int strideT = gridDim.x;
    const int maxDw = F * 3 - 1;        // clamp bound for tail tiles

    // issue one tile: 6 async b32 per thread, each instruction a contiguous
    // 128B/wave transaction (lane t copies dwords t, t+256, t+512)
    auto issue_tile = [&](int tile, int bufsel) {
        const int base = tile * TILE_DW;
        #pragma unroll
        for (int k = 0; k < 3; ++k) {
            int j = t + k * TILE;
            int a = base + j;
            a = (a > maxDw) ? maxDw : a;                 // stay in-bounds, keep EXEC full
            async_ld_b32(lds_off_of(&facesLDS[bufsel][j]), faces + a);
        }
        #pragma unroll
        for (int k = 0; k < 3; ++k) {
            int j = t + k * TILE;
            int a = base + j;
            a = (a > maxDw) ? maxDw : a;
            async_ld_b32(lds_off_of(&uvLDS[bufsel][j]), uv_faces + a);
        }
    };

    int tile = blockIdx.x;
    if (tile >= numTiles) return;

    issue_tile(tile, 0);
    int cur = 0;

    for (; tile < numTiles; tile += strideT) {
        const int nextTile = tile + strideT;
        const bool hasNext = nextTile < numTiles;

        // L2 warm-up two tiles ahead (gfx1250 global_prefetch_b8)
        int pfTile = tile + 2 * strideT;
        if (pfTile < numTiles) {
            const char* pf = (const char*)(faces + (size_t)pfTile * TILE_DW);
            const char* pu = (const char*)(uv_faces + (size_t)pfTile * TILE_DW);
            if (t < 24)      __builtin_prefetch(pf + t * 128, 0, 1);
            else if (t < 48) __builtin_prefetch(pu + (t - 24) * 128, 0, 1);
        }

        // pipeline: launch next tile's copies, then wait for *current* tile
        if (hasNext) { issue_tile(nextTile, cur ^ 1); wait_async6(); }
        else         { wait_async0(); }
        __syncthreads();   // LDS tile visible to all waves

        const int fi = tile * TILE + t;
        if (fi < F) {
            const int b = 3 * t;
            const int fx = facesLDS[cur][b + 0];
            const int fy = facesLDS[cur][b + 1];
            const int fz = facesLDS[cur][b + 2];

            V3f p0 = *(const V3f*)(positions + 3 * fx);
            V3f p1 = *(const V3f*)(positions + 3 * fy);
            V3f p2 = *(const V3f*)(positions + 3 * fz);

            float e1x = p1.x - p0.x, e1y = p1.y - p0.y, e1z = p1.z - p0.z;
            float e2x = p2.x - p0.x, e2y = p2.y - p0.y, e2z = p2.z - p0.z;

            float fnx = e1y * e2z - e1z * e2y;
            float fny = e1z * e2x - e1x * e2z;
            float fnz = e1x * e2y - e1y * e2x;

            gatomic_add_f32(&nrm_accum[3 * fx + 0], fnx);
            gatomic_add_f32(&nrm_accum[3 * fx + 1], fny);
            gatomic_add_f32(&nrm_accum[3 * fx + 2], fnz);
            gatomic_add_f32(&nrm_accum[3 * fy + 0], fnx);
            gatomic_add_f32(&nrm_accum[3 * fy + 1], fny);
            gatomic_add_f32(&nrm_accum[3 * fy + 2], fnz);
            gatomic_add_f32(&nrm_accum[3 * fz + 0], fnx);
            gatomic_add_f32(&nrm_accum[3 * fz + 1], fny);
            gatomic_add_f32(&nrm_accum[3 * fz + 2], fnz);

            const int ux = uvLDS[cur][b + 0];
            const int uy = uvLDS[cur][b + 1];
            const int uz = uvLDS[cur][b + 2];

            V2f t0 = *(const V2f*)(texcoords + 2 * ux);
            V2f t1 = *(const V2f*)(texcoords + 2 * uy);
            V2f t2 = *(const V2f*)(texcoords + 2 * uz);

            float u1x = t1.x - t0.x, u1y = t1.y - t0.y;
            float u2x = t2.x - t0.x, u2y = t2.y - t0.y;

            float nomx = e1x * u2y - e2x * u1y;
            float nomy = e1y * u2y - e2y * u1y;
            float nomz = e1z * u2y - e2z * u1y;

            float denom = u1x * u2y - u1y * u2x;
            denom = (denom > 0.0f) ? fmaxf(denom, EPS_DENOM) : fminf(denom, -EPS_DENOM);

            float tgx = nomx / denom;
            float tgy = nomy / denom;
            float tgz = nomz / denom;

            gatomic_add_f32(&tan_accum[3 * fx + 0], tgx);
            gatomic_add_f32(&tan_accum[3 * fx + 1], tgy);
            gatomic_add_f32(&tan_accum[3 * fx + 2], tgz);
            gatomic_add_f32(&tan_accum[3 * fy + 0], tgx);
            gatomic_add_f32(&tan_accum[3 * fy + 1], tgy);
            gatomic_add_f32(&tan_accum[3 * fy + 2], tgz);
            gatomic_add_f32(&tan_accum[3 * fz + 0], tgx);
            gatomic_add_f32(&tan_accum[3 * fz + 1], tgy);
            gatomic_add_f32(&tan_accum[3 * fz + 2], tgz);
        }

        __syncthreads();   // everyone done with buf[cur] before it is refilled
        cur ^= 1;
    }
}

// Vertex-parallel finalize, in place on d_out (exact reference semantics).
__global__ __launch_bounds__(256) void vg_vertex_kernel(float* __restrict__ out, int V) {
    int v = blockIdx.x * blockDim.x + threadIdx.x;
    if (v >= V) return;

    float* np = out + 3 * (size_t)v;
    float* tp = out + 3 * (size_t)V + 3 * (size_t)v;

    V3f n = *(const V3f*)np;
    float dn = n.x * n.x + n.y * n.y + n.z * n.z;
    if (!(dn > EPS_LEN)) { n.x = 0.f; n.y = 0.f; n.z = 1.f; dn = 1.f; }
    float invn = 1.0f / sqrtf(fmaxf(dn, EPS_LEN));
    n.x *= invn; n.y *= invn; n.z *= invn;

    V3f t = *(const V3f*)tp;
    float dt = t.x * t.x + t.y * t.y + t.z * t.z;
    float invt = 1.0f / sqrtf(fmaxf(dt, EPS_LEN));
    t.x *= invt; t.y *= invt; t.z *= invt;

    float tn = t.x * n.x + t.y * n.y + t.z * n.z;
    t.x -= tn * n.x; t.y -= tn * n.y; t.z -= tn * n.z;

    float dt2 = t.x * t.x + t.y * t.y + t.z * t.z;
    float invt2 = 1.0f / sqrtf(fmaxf(dt2, EPS_LEN));
    t.x *= invt2; t.y *= invt2; t.z *= invt2;

    *(V3f*)np = n;
    *(V3f*)tp = t;
}

extern "C" void kernel_launch(void* const* d_in, const int* in_sizes, int n_in,
                              void* d_out, int out_size, void* d_ws, size_t ws_size,
                              hipStream_t stream) {
    const float* positions = (const float*)d_in[0];
    const int*   faces     = (const int*)d_in[1];
    const float* texcoords = (const float*)d_in[2];
    const int*   uv_faces  = (const int*)d_in[3];

    const int V = in_sizes[0] / 3;
    const int F = in_sizes[1] / 3;
    (void)n_in; (void)d_ws; (void)ws_size;

    float* out = (float*)d_out;
    float* nrm_accum = out;                    // [V,3]
    float* tan_accum = out + (size_t)V * 3;    // [V,3]

    // 1) zero the accumulators
    const int n4 = out_size / 4;
    {
        int blocks = (n4 + 255) / 256;
        if (blocks < 1) blocks = 1;
        vg_zero_kernel<<<blocks, 256, 0, stream>>>((float4*)d_out, n4, out, out_size);
    }

    // 2) face-parallel scatter: persistent blocks, double-buffered async tiles
    const int numTiles = (F + TILE - 1) / TILE;
    int faceBlocks = numTiles < 4096 ? numTiles : 4096;
    if (faceBlocks < 1) faceBlocks = 1;
    vg_face_kernel<<<faceBlocks, TILE, 0, stream>>>(
        positions, faces, texcoords, uv_faces, nrm_accum, tan_accum, F, numTiles);

    // 3) vertex-parallel normalization / orthogonalization, in place
    vg_vertex_kernel<<<(V + 255) / 256, 256, 0, stream>>>(out, V);
}